// VisionMoBA_9457517986198
// MI455X (gfx1250) — compile-verified
//
#include <hip/hip_runtime.h>
#include <hip/hip_bf16.h>

// MoBA attention forward for MI455X (gfx1250), wave32, f32 WMMA 16x16x4,
// async-to-LDS (ASYNCcnt) staging with double buffering in the GEMMs.

#define S_LEN 2048
#define HID 2048
#define NHEAD 16
#define HDIM 128
#define NBLK 32
#define BLKSZ 64
#define NEG_BIG (-1.0e30f)
#define POS_BIG (1.0e30f)

typedef float v2f __attribute__((ext_vector_type(2)));
typedef float v4f __attribute__((ext_vector_type(4)));
typedef float v8f __attribute__((ext_vector_type(8)));

static __device__ __forceinline__ v8f wmma4(v2f a, v2f b, v8f c) {
    // D = A(16x4,f32) * B(4x16,f32) + C(16x16,f32)
    return __builtin_amdgcn_wmma_f32_16x16x4_f32(
        /*neg_a=*/false, a, /*neg_b=*/false, b,
        /*c_mod=*/(short)0, c, /*reuse_a=*/false, /*reuse_b=*/false);
}

static __device__ __forceinline__ v8f v8f_zero() {
    v8f z;
#pragma unroll
    for (int e = 0; e < 8; ++e) z[e] = 0.0f;
    return z;
}

// --- CDNA5 async global->LDS DMA (ASYNCcnt), no VGPR round-trip ------------
static __device__ __forceinline__ unsigned lds_off(const void* p) {
    // Low 32 bits of a generic LDS pointer are the wave-relative LDS offset
    // (shared aperture base has zero low bits).
    return (unsigned)(size_t)p;
}
static __device__ __forceinline__ void async_ld_b128(unsigned lds, const float* g) {
    asm volatile("global_load_async_to_lds_b128 %0, %1, off"
                 :: "v"(lds), "v"(g) : "memory");
}
static __device__ __forceinline__ void async_ld_b32(unsigned lds, const float* g) {
    asm volatile("global_load_async_to_lds_b32 %0, %1, off"
                 :: "v"(lds), "v"(g) : "memory");
}
static __device__ __forceinline__ void wait_async_le10() {
    asm volatile("s_wait_asynccnt 0xa" ::: "memory");
}
static __device__ __forceinline__ void wait_async_le0() {
    asm volatile("s_wait_asynccnt 0x0" ::: "memory");
}

// ---------------------------------------------------------------------------
// Generic f32 GEMM: C[M,N] = A[M,K] @ B[K,N], all row-major, dims % 64 == 0.
// Block: 128 threads = 4 waves; block tile 64x64; each wave 32x32 (2x2 WMMA).
// K staged 16-wide, double-buffered via async-to-LDS DMA (10 ops/thread/tile:
// 2x b128 for A row-major, 8x b32 for B transposed scatter). Async loads
// retire in order, so s_wait_asynccnt<=10 drains the previous tile while the
// next tile's DMA overlaps the WMMA compute phase.
// ---------------------------------------------------------------------------
__global__ __launch_bounds__(128) void gemm_f32(const float* __restrict__ A,
                                                const float* __restrict__ B,
                                                float* __restrict__ C,
                                                int M, int N, int K) {
    __shared__ float sA[2][64 * 16];  // [m][k]
    __shared__ float sB[2][64 * 16];  // [n][k]  (B tile stored transposed)

    const int tid  = threadIdx.x;
    const int lane = tid & 31;
    const int w    = tid >> 5;
    const int lrow = lane & 15;
    const int lhi  = lane >> 4;

    const int row0 = blockIdx.y * 64;
    const int col0 = blockIdx.x * 64;
    const int wr = (w >> 1) * 32;
    const int wc = (w & 1) * 32;

    // Per-thread staging addresses
    const int mA = tid >> 1, k0A = (tid & 1) * 8;
    const float* gA = &A[(size_t)(row0 + mA) * K + k0A];
    const int kkB = tid >> 3, n0B = (tid & 7) * 8;
    const float* gB = &B[(size_t)kkB * N + col0 + n0B];

    unsigned ldsA[2], ldsB[2];
    ldsA[0] = lds_off(&sA[0][mA * 16 + k0A]);
    ldsA[1] = lds_off(&sA[1][mA * 16 + k0A]);
    ldsB[0] = lds_off(&sB[0][n0B * 16 + kkB]);
    ldsB[1] = lds_off(&sB[1][n0B * 16 + kkB]);

    auto stage = [&](int buf, int kt) {
        async_ld_b128(ldsA[buf],      gA + kt);
        async_ld_b128(ldsA[buf] + 16, gA + kt + 4);
        const float* gb = gB + (size_t)kt * N;
#pragma unroll
        for (int qq = 0; qq < 8; ++qq)
            async_ld_b32(ldsB[buf] + qq * (16 * 4), gb + qq);
    };

    v8f acc[2][2];
#pragma unroll
    for (int i = 0; i < 2; ++i)
#pragma unroll
        for (int j = 0; j < 2; ++j) acc[i][j] = v8f_zero();

    stage(0, 0);
    const int nk = K >> 4;
    for (int i = 0; i < nk; ++i) {
        const int c = i & 1;
        if (i + 1 < nk) {
            stage(c ^ 1, (i + 1) * 16);  // DMA next tile during compute
            wait_async_le10();           // previous tile fully landed
        } else {
            wait_async_le0();
        }
        __syncthreads();                 // everyone's tile visible

#pragma unroll
        for (int st = 0; st < 4; ++st) {
            const int kloc = st * 4 + 2 * lhi;
            v2f a0 = *(const v2f*)&sA[c][(wr + lrow) * 16 + kloc];
            v2f a1 = *(const v2f*)&sA[c][(wr + 16 + lrow) * 16 + kloc];
            v2f b0 = *(const v2f*)&sB[c][(wc + lrow) * 16 + kloc];
            v2f b1 = *(const v2f*)&sB[c][(wc + 16 + lrow) * 16 + kloc];
            acc[0][0] = wmma4(a0, b0, acc[0][0]);
            acc[0][1] = wmma4(a0, b1, acc[0][1]);
            acc[1][0] = wmma4(a1, b0, acc[1][0]);
            acc[1][1] = wmma4(a1, b1, acc[1][1]);
        }
        __syncthreads();                 // safe to overwrite this buffer later
    }

#pragma unroll
    for (int mi = 0; mi < 2; ++mi)
#pragma unroll
        for (int ni = 0; ni < 2; ++ni)
#pragma unroll
            for (int r = 0; r < 8; ++r) {
                const int m = row0 + wr + mi * 16 + 8 * lhi + r;
                const int n = col0 + wc + ni * 16 + lrow;
                C[(size_t)m * N + n] = acc[mi][ni][r];
            }
}

// ---------------------------------------------------------------------------
// kg[h][d][j] = mean over t in block j of K[j*64+t][h*128+d]
// ---------------------------------------------------------------------------
__global__ __launch_bounds__(256) void kg_mean(const float* __restrict__ k,
                                               float* __restrict__ kg) {
    const int idx = blockIdx.x * 256 + threadIdx.x;  // H*NBLK*HDIM = 65536
    const int d = idx & 127;
    const int j = (idx >> 7) & 31;
    const int h = idx >> 12;
    float s = 0.f;
    for (int t = 0; t < BLKSZ; ++t)
        s += k[(size_t)(j * BLKSZ + t) * HID + h * HDIM + d];
    kg[(h * HDIM + d) * NBLK + j] = s * (1.0f / BLKSZ);
}

// ---------------------------------------------------------------------------
// One wave per (h,s); lane j owns block j. Computes gate, masks, top-8,
// writes additive bias[h][s][j] in {0, NEG_BIG}.
// ---------------------------------------------------------------------------
__global__ __launch_bounds__(256) void gate_topk(const float* __restrict__ q,
                                                 const float* __restrict__ kg,
                                                 float* __restrict__ bias) {
    const int lane = threadIdx.x & 31;
    const int wid  = blockIdx.x * 8 + (threadIdx.x >> 5);  // 0 .. H*S-1
    const int h = wid >> 11;
    const int s = wid & (S_LEN - 1);

    const float* qrow = q + (size_t)s * HID + h * HDIM;
    const float* kgp  = kg + (size_t)(h * HDIM) * NBLK + lane;
    float acc = 0.f;
    for (int d = 0; d < HDIM; ++d) acc = fmaf(qrow[d], kgp[d * NBLK], acc);

    const int qb = s >> 6;
    float g;
    if (lane == qb)                  g = POS_BIG;  // self block: always selected
    else if (s < (lane + 1) * BLKSZ) g = NEG_BIG;  // not fully in the past
    else                             g = acc;

    bool sel = false;
#pragma unroll
    for (int it = 0; it < 8; ++it) {
        float cand = sel ? -3.0e38f : g;
        float m = cand;
#pragma unroll
        for (int msk = 16; msk >= 1; msk >>= 1)
            m = fmaxf(m, __shfl_xor(m, msk, 32));
        unsigned long long bal = __ballot(cand == m);
        int first = __ffsll(bal) - 1;
        if (lane == first) sel = true;
    }
    bias[(size_t)wid * NBLK + lane] = sel ? 0.0f : NEG_BIG;
}

// ---------------------------------------------------------------------------
// Flash-style MoBA attention. Workgroup = (head, 64-row query block):
// 4 waves x 16 query rows. K_j staged [t][d] via async DMA; V_j re-staged
// transposed [d][t] in the same LDS buffer; P converted C-layout -> A-layout
// via per-wave LDS.
// ---------------------------------------------------------------------------
__global__ __launch_bounds__(128) void moba_attn(const float* __restrict__ q,
                                                 const float* __restrict__ k,
                                                 const float* __restrict__ v,
                                                 const float* __restrict__ bias,
                                                 float* __restrict__ o) {
    __shared__ float sKV[BLKSZ * HDIM];  // 32 KB, K then V(transposed)
    __shared__ float sP[4][16 * BLKSZ];  // per-wave P tile, 16 KB

    const int tid  = threadIdx.x;
    const int lane = tid & 31;
    const int w    = tid >> 5;
    const int lrow = lane & 15;
    const int lhi  = lane >> 4;

    const int h  = blockIdx.y;
    const int qb = blockIdx.x;
    const int s0 = qb * BLKSZ + w * 16;   // this wave's first query row
    const float sm_scale = 0.08838834764831845f;  // 1/sqrt(128)

    // Q fragments (A-operand layout): lane holds Q[s0+lrow][4*st + 2*lhi + {0,1}]
    v2f qf[32];
    {
        const float* qp = q + (size_t)(s0 + lrow) * HID + h * HDIM + 2 * lhi;
#pragma unroll
        for (int st = 0; st < 32; ++st) qf[st] = *(const v2f*)&qp[st * 4];
    }

    float mstat[8], lstat[8];
    v8f oacc[8];
#pragma unroll
    for (int r = 0; r < 8; ++r) { mstat[r] = NEG_BIG; lstat[r] = 0.f; }
#pragma unroll
    for (int dt = 0; dt < 8; ++dt) oacc[dt] = v8f_zero();

    const unsigned sKV_base = lds_off(&sKV[0]);

    for (int j = 0; j <= qb; ++j) {
        // ---- stage K_j as [t][d] via async DMA ----
        __syncthreads();
        {
            const float* gk = &k[(size_t)(j * BLKSZ) * HID + h * HDIM];
#pragma unroll
            for (int c = 0; c < 16; ++c) {
                const int id = c * 128 + tid;
                const int tr = id >> 5;
                const int d4 = (id & 31) * 4;
                async_ld_b128(sKV_base + (unsigned)(tr * HDIM + d4) * 4,
                              gk + (size_t)tr * HID + d4);
            }
            wait_async_le0();
        }
        __syncthreads();

        // ---- S = Q K_j^T  (16 x 64), 4 column tiles ----
        v8f sacc[4];
#pragma unroll
        for (int nt = 0; nt < 4; ++nt) sacc[nt] = v8f_zero();
#pragma unroll
        for (int nt = 0; nt < 4; ++nt) {
            const float* bp = &sKV[(nt * 16 + lrow) * HDIM + 2 * lhi];
#pragma unroll
            for (int st = 0; st < 32; ++st) {
                v2f b = *(const v2f*)&bp[st * 4];
                sacc[nt] = wmma4(qf[st], b, sacc[nt]);
            }
        }

        // ---- bias / causal mask / scale ----
        float bval[8];
        if (j < qb) {
#pragma unroll
            for (int r = 0; r < 8; ++r) {
                const int srow = s0 + 8 * lhi + r;
                bval[r] = bias[((size_t)(h * S_LEN + srow)) * NBLK + j];
            }
        } else {
#pragma unroll
            for (int r = 0; r < 8; ++r) bval[r] = 0.f;
        }
#pragma unroll
        for (int nt = 0; nt < 4; ++nt) {
#pragma unroll
            for (int r = 0; r < 8; ++r) {
                float vv = sacc[nt][r] * sm_scale + bval[r];
                if (j == qb) {
                    const int t = j * BLKSZ + nt * 16 + lrow;
                    const int srow = s0 + 8 * lhi + r;
                    if (t > srow) vv = NEG_BIG;
                }
                sacc[nt][r] = vv;
            }
        }

        // ---- online softmax update (rows live in C layout: 8 rows/lane) ----
#pragma unroll
        for (int r = 0; r < 8; ++r) {
            float rmax = sacc[0][r];
#pragma unroll
            for (int nt = 1; nt < 4; ++nt) rmax = fmaxf(rmax, sacc[nt][r]);
#pragma unroll
            for (int msk = 8; msk >= 1; msk >>= 1)
                rmax = fmaxf(rmax, __shfl_xor(rmax, msk, 32));

            const float newm = fmaxf(mstat[r], rmax);
            const float corr = __expf(mstat[r] - newm);
            mstat[r] = newm;

            float rs = 0.f;
#pragma unroll
            for (int nt = 0; nt < 4; ++nt) {
                const float p = __expf(sacc[nt][r] - newm);
                sP[w][(8 * lhi + r) * BLKSZ + nt * 16 + lrow] = p;
                rs += p;
            }
#pragma unroll
            for (int msk = 8; msk >= 1; msk >>= 1)
                rs += __shfl_xor(rs, msk, 32);

            lstat[r] = lstat[r] * corr + rs;
#pragma unroll
            for (int dt = 0; dt < 8; ++dt) oacc[dt][r] *= corr;
        }

        // ---- stage V_j transposed [d][t] (reusing sKV) ----
        __syncthreads();
#pragma unroll
        for (int c = 0; c < 16; ++c) {
            const int id = c * 128 + tid;
            const int tr = id >> 5;
            const int d4 = (id & 31) * 4;
            v4f vv = *(const v4f*)&v[(size_t)(j * BLKSZ + tr) * HID + h * HDIM + d4];
            sKV[(d4 + 0) * BLKSZ + tr] = vv[0];
            sKV[(d4 + 1) * BLKSZ + tr] = vv[1];
            sKV[(d4 + 2) * BLKSZ + tr] = vv[2];
            sKV[(d4 + 3) * BLKSZ + tr] = vv[3];
        }
        __syncthreads();

        // ---- O += P @ V_j  (16 x 128), 8 d tiles ----
        v2f pf[16];
        {
            const float* ap = &sP[w][lrow * BLKSZ + 2 * lhi];
#pragma unroll
            for (int st = 0; st < 16; ++st) pf[st] = *(const v2f*)&ap[st * 4];
        }
#pragma unroll
        for (int dt = 0; dt < 8; ++dt) {
            const float* bp = &sKV[(dt * 16 + lrow) * BLKSZ + 2 * lhi];
#pragma unroll
            for (int st = 0; st < 16; ++st) {
                v2f b = *(const v2f*)&bp[st * 4];
                oacc[dt] = wmma4(pf[st], b, oacc[dt]);
            }
        }
    }

    // ---- epilogue: normalize and store o[s][h*128 + d] ----
    float inv[8];
#pragma unroll
    for (int r = 0; r < 8; ++r) inv[r] = 1.0f / lstat[r];
#pragma unroll
    for (int dt = 0; dt < 8; ++dt)
#pragma unroll
        for (int r = 0; r < 8; ++r) {
            const int srow = s0 + 8 * lhi + r;
            const int d = dt * 16 + lrow;
            o[(size_t)srow * HID + h * HDIM + d] = oacc[dt][r] * inv[r];
        }
}

// ---------------------------------------------------------------------------
extern "C" void kernel_launch(void* const* d_in, const int* in_sizes, int n_in,
                              void* d_out, int out_size, void* d_ws, size_t ws_size,
                              hipStream_t stream) {
    const float* X  = (const float*)d_in[0];
    const float* Wq = (const float*)d_in[1];
    const float* Wk = (const float*)d_in[2];
    const float* Wv = (const float*)d_in[3];
    const float* Wo = (const float*)d_in[4];
    float* out = (float*)d_out;

    float* base = (float*)d_ws;
    const size_t MAT = (size_t)S_LEN * HID;  // 2048*2048
    float* q    = base;
    float* k    = base + 1 * MAT;
    float* v    = base + 2 * MAT;
    float* o    = base + 3 * MAT;
    float* kg   = base + 4 * MAT;                        // H*D*NBLK = 65536
    float* bias = kg + (size_t)NHEAD * HDIM * NBLK;      // H*S*NBLK = 1M

    dim3 gemm_grid(HID / 64, S_LEN / 64);
    gemm_f32<<<gemm_grid, 128, 0, stream>>>(X, Wq, q, S_LEN, HID, HID);
    gemm_f32<<<gemm_grid, 128, 0, stream>>>(X, Wk, k, S_LEN, HID, HID);
    gemm_f32<<<gemm_grid, 128, 0, stream>>>(X, Wv, v, S_LEN, HID, HID);

    kg_mean<<<(NHEAD * NBLK * HDIM) / 256, 256, 0, stream>>>(k, kg);
    gate_topk<<<(NHEAD * S_LEN) / 8, 256, 0, stream>>>(q, kg, bias);

    moba_attn<<<dim3(S_LEN / BLKSZ, NHEAD), 128, 0, stream>>>(q, k, v, bias, o);

    gemm_f32<<<gemm_grid, 128, 0, stream>>>(o, Wo, out, S_LEN, HID, HID);
}